// HGTTransformerModel_38500086841761
// MI455X (gfx1250) — compile-verified
//
#include <hip/hip_runtime.h>
#include <math.h>

// ---------------------------------------------------------------------------
// HGT + TimeTransformer forward for MI455X (gfx1250, wave32).
// Dense GEMMs run on V_WMMA_F32_16X16X4_F32 (fp32 matrix pipe, exact dtype
// match with the fp32 reference). Graph segment-softmax uses atomics; time
// attention is a flash-style LDS two-pass per (query, head).
//
// Assumed input flattening (setup_inputs() insertion order, params recursive):
//  0 node_x[50000,16] 1 time_x[2048,8] 2 edge_nn[2,800000] 3 edge_bt[2,50000]
//  4 proj.node.W 5 proj.node.b 6 proj.time.W 7 proj.time.b
//  conv[L] base = 8 + 24*L:
//   +0 k.node.W +1 k.node.b +2 k.time.W +3 k.time.b
//   +4 q.node.W +5 q.node.b +6 q.time.W +7 q.time.b
//   +8 v.node.W +9 v.node.b +10 v.time.W +11 v.time.b
//   +12 a.node.W +13 a.node.b +14 a.time.W +15 a.time.b
//   +16 skip.node +17 skip.time
//   +18 rel.nn.a_rel +19 rel.nn.m_rel +20 rel.nn.p_rel
//   +21 rel.bt.a_rel +22 rel.bt.m_rel +23 rel.bt.p_rel
//  56 tin.W 57 tin.b
//  enc[L] base = 58 + 12*L:
//   +0 qkv_W +1 qkv_b +2 out.W +3 out.b +4 ff1.W +5 ff1.b +6 ff2.W +7 ff2.b
//   +8 ln1_g +9 ln1_b +10 ln2_g +11 ln2_b
//  82 head.W 83 head.b 84 dec1.W 85 dec1.b 86 dec2.W 87 dec2.b
// ---------------------------------------------------------------------------

#define HID   128
#define NHEAD 4
#define DHEAD 32
#define NNODE 50000
#define NTOK  2048
#define ENN   800000

typedef __attribute__((ext_vector_type(2))) float v2f;
typedef __attribute__((ext_vector_type(8))) float v8f;

#define RSQRT_D 0.17677669529663687f   // 1/sqrt(32)

// ------------------------------- WMMA GEMM ---------------------------------
// C[M,N] = A[M,K] @ W[K,N] (+bias) (+relu). Wave computes 16 x (NT*16) strip.
// Requires M%16==0, K%4==0, N multiple of NT*16 (guaranteed by launcher).
// f32 16x16x4 fragment layouts per CDNA5 ISA 7.12.2:
//   A lane L: row M = L%16, VGPR r holds K = kbase + r + 2*(L/16)
//   B lane L: col N = L%16, VGPR r holds K = kbase + r + 2*(L/16)
//   C lane L: VGPR r -> row = r + 8*(L/16), col = L%16
template <int NT>
__global__ void gemm_wmma_f32(const float* __restrict__ A, int lda, long sAb,
                              const float* __restrict__ W, int ldw, long sWb,
                              const float* __restrict__ bias,
                              float* __restrict__ C, int ldc, long sCb,
                              int K, int act) {
  const int tm  = blockIdx.x * 16;
  const int tn0 = blockIdx.y * (NT * 16);
  A += (long)blockIdx.z * sAb;
  W += (long)blockIdx.z * sWb;
  C += (long)blockIdx.z * sCb;

  const int lane = threadIdx.x & 31;
  const int half = lane >> 4;
  const int l    = lane & 15;

  v8f acc[NT];
#pragma unroll
  for (int j = 0; j < NT; ++j) acc[j] = (v8f){0.f,0.f,0.f,0.f,0.f,0.f,0.f,0.f};

  const float* arow = A + (long)(tm + l) * lda;
  for (int k = 0; k < K; k += 4) {
    const int ka = k + 2 * half;
    v2f afrag;
    afrag.x = arow[ka];
    afrag.y = arow[ka + 1];
#pragma unroll
    for (int j = 0; j < NT; ++j) {
      const float* bp = W + (long)ka * ldw + tn0 + j * 16 + l;
      v2f bfrag;
      bfrag.x = bp[0];
      bfrag.y = bp[ldw];
      acc[j] = __builtin_amdgcn_wmma_f32_16x16x4_f32(
          false, afrag, false, bfrag, (short)0, acc[j], false, false);
    }
  }

#pragma unroll
  for (int j = 0; j < NT; ++j) {
    const int col = tn0 + j * 16 + l;
    const float bv = bias ? bias[col] : 0.0f;
#pragma unroll
    for (int r = 0; r < 8; ++r) {
      const int row = tm + r + 8 * half;
      float v = acc[j][r] + bv;
      if (act == 1) v = v > 0.f ? v : 0.f;
      C[(long)row * ldc + col] = v;
    }
  }
}

// ------------------------------ elementwise --------------------------------
__global__ void fill_f32(float* __restrict__ p, float v, long n) {
  long i = (long)blockIdx.x * blockDim.x + threadIdx.x;
  if (i < n) p[i] = v;
}

__global__ void gelu_ip(float* __restrict__ x, long n) {
  long i = (long)blockIdx.x * blockDim.x + threadIdx.x;
  if (i < n) {
    float v = x[i];
    float c = 0.7978845608028654f * (v + 0.044715f * v * v * v);
    x[i] = 0.5f * v * (1.f + tanhf(c));
  }
}

__global__ void skip_relu(const float* __restrict__ o, const float* __restrict__ xin,
                          const float* __restrict__ skp, float* __restrict__ xout, long n) {
  long i = (long)blockIdx.x * blockDim.x + threadIdx.x;
  if (i < n) {
    float beta = 1.f / (1.f + __expf(-skp[0]));
    float v = beta * o[i] + (1.f - beta) * xin[i];
    xout[i] = v > 0.f ? v : 0.f;
  }
}

// ----------------------------- edge softmax --------------------------------
__device__ inline void atomicMaxF(float* a, float v) {
  if (v >= 0.f) atomicMax((int*)a, __float_as_int(v));
  else          atomicMin((unsigned int*)a, (unsigned int)__float_as_int(v));
}

__global__ void edge_alpha(const int* __restrict__ src, const int* __restrict__ dst,
                           const float* __restrict__ q, const float* __restrict__ krel,
                           const float* __restrict__ prel,
                           float* __restrict__ alpha, float* __restrict__ amax, int E) {
  long tid = (long)blockIdx.x * blockDim.x + threadIdx.x;
  if (tid >= (long)E * NHEAD) return;
  int e = (int)(tid >> 2), h = (int)(tid & 3);
  int s = src[e], d = dst[e];
  const float* qp = q    + (long)d * HID + h * DHEAD;
  const float* kp = krel + (long)s * HID + h * DHEAD;
  float acc = 0.f;
#pragma unroll
  for (int i = 0; i < DHEAD; ++i) acc += qp[i] * kp[i];
  float a = acc * prel[h] * RSQRT_D;
  alpha[tid] = a;
  atomicMaxF(&amax[(long)d * NHEAD + h], a);
}

__global__ void edge_expsum(const int* __restrict__ dst, float* __restrict__ alpha,
                            const float* __restrict__ amax, float* __restrict__ den, int E) {
  long tid = (long)blockIdx.x * blockDim.x + threadIdx.x;
  if (tid >= (long)E * NHEAD) return;
  int e = (int)(tid >> 2), h = (int)(tid & 3);
  int d = dst[e];
  float ex = __expf(alpha[tid] - amax[(long)d * NHEAD + h]);
  alpha[tid] = ex;
  atomicAdd(&den[(long)d * NHEAD + h], ex);
}

__global__ void edge_agg(const int* __restrict__ src, const int* __restrict__ dst,
                         const float* __restrict__ alpha, const float* __restrict__ den,
                         const float* __restrict__ vrel, float* __restrict__ out, int E) {
  long tid = (long)blockIdx.x * blockDim.x + threadIdx.x;
  if (tid >= (long)E * NHEAD) return;
  int e = (int)(tid >> 2), h = (int)(tid & 3);
  int s = src[e], d = dst[e];
  float dn = den[(long)d * NHEAD + h];
  dn = dn > 1e-16f ? dn : 1e-16f;
  float w = alpha[tid] / dn;
  const float* vp = vrel + (long)s * HID + h * DHEAD;
  float* op       = out  + (long)d * HID + h * DHEAD;
#pragma unroll
  for (int i = 0; i < DHEAD; ++i) atomicAdd(&op[i], w * vp[i]);
}

// ------------------------------ scatter mean -------------------------------
__global__ void scatter_sum(const int* __restrict__ src, const int* __restrict__ dst,
                            const float* __restrict__ hn, float* __restrict__ te,
                            float* __restrict__ cnt, int E) {
  long tid = (long)blockIdx.x * blockDim.x + threadIdx.x;
  if (tid >= (long)E * HID) return;
  int e = (int)(tid >> 7), d = (int)(tid & 127);
  int s = src[e], t = dst[e];
  atomicAdd(&te[(long)t * HID + d], hn[(long)s * HID + d]);
  if (d == 0) atomicAdd(&cnt[t], 1.f);
}

__global__ void scatter_div(float* __restrict__ te, const float* __restrict__ cnt) {
  long tid = (long)blockIdx.x * blockDim.x + threadIdx.x;
  if (tid >= (long)NTOK * HID) return;
  float c = cnt[tid >> 7];
  te[tid] /= (c == 0.f ? 1.f : c);
}

// ----------------------------- time attention ------------------------------
// one block per (query, head); two-pass softmax with scores staged in LDS
__global__ void mha_enc(const float* __restrict__ qkv, float* __restrict__ o) {
  __shared__ float sc[NTOK];
  __shared__ float rbuf[256];
  __shared__ float pm[8][DHEAD];
  const int q = blockIdx.x, h = blockIdx.y, t = threadIdx.x;
  const float* qrow = qkv + (long)q * 384 + h * DHEAD;

  float lmax = -3.0e38f;
  for (int k = t; k < NTOK; k += 256) {
    const float* krow = qkv + (long)k * 384 + 128 + h * DHEAD;
    float d = 0.f;
#pragma unroll
    for (int i = 0; i < DHEAD; ++i) d += qrow[i] * krow[i];
    d *= RSQRT_D;
    sc[k] = d;
    lmax = fmaxf(lmax, d);
  }
  rbuf[t] = lmax; __syncthreads();
  for (int s = 128; s > 0; s >>= 1) { if (t < s) rbuf[t] = fmaxf(rbuf[t], rbuf[t + s]); __syncthreads(); }
  const float M = rbuf[0]; __syncthreads();

  float lsum = 0.f;
  for (int k = t; k < NTOK; k += 256) { float e = __expf(sc[k] - M); sc[k] = e; lsum += e; }
  rbuf[t] = lsum; __syncthreads();
  for (int s = 128; s > 0; s >>= 1) { if (t < s) rbuf[t] += rbuf[t + s]; __syncthreads(); }
  const float S = rbuf[0]; __syncthreads();

  const int d  = t & 31;
  const int sl = t >> 5;     // 8 key slices
  float pa = 0.f;
  for (int k = sl; k < NTOK; k += 8)
    pa += sc[k] * qkv[(long)k * 384 + 256 + h * DHEAD + d];
  pm[sl][d] = pa; __syncthreads();
  if (t < DHEAD) {
    float s2 = 0.f;
#pragma unroll
    for (int i = 0; i < 8; ++i) s2 += pm[i][t];
    o[(long)q * HID + h * DHEAD + t] = s2 / S;
  }
}

// x = LN(x + r) * g + b   (block = one row, 128 threads)
__global__ void ln_residual(float* __restrict__ x, const float* __restrict__ r,
                            const float* __restrict__ g, const float* __restrict__ b) {
  __shared__ float rb[HID];
  const int row = blockIdx.x, t = threadIdx.x;
  float v = x[(long)row * HID + t] + r[(long)row * HID + t];
  rb[t] = v; __syncthreads();
  for (int s = 64; s > 0; s >>= 1) { if (t < s) rb[t] += rb[t + s]; __syncthreads(); }
  const float mean = rb[0] * (1.f / HID); __syncthreads();
  const float dv = v - mean;
  rb[t] = dv * dv; __syncthreads();
  for (int s = 64; s > 0; s >>= 1) { if (t < s) rb[t] += rb[t + s]; __syncthreads(); }
  const float var = rb[0] * (1.f / HID);
  x[(long)row * HID + t] = dv * rsqrtf(var + 1e-5f) * g[t] + b[t];
}

// --------------------------- head + tiny decoder ---------------------------
__global__ void head_dec(const float* __restrict__ x,
                         const float* __restrict__ hW, const float* __restrict__ hb,
                         const float* __restrict__ d1W, const float* __restrict__ d1b,
                         const float* __restrict__ d2W, const float* __restrict__ d2b,
                         float* __restrict__ out) {
  int t = blockIdx.x * blockDim.x + threadIdx.x;
  if (t >= NTOK) return;
  float p = hb[0];
#pragma unroll 4
  for (int i = 0; i < HID; ++i) p += x[(long)t * HID + i] * hW[i];
  float o = d2b[0];
#pragma unroll
  for (int j = 0; j < 32; ++j) {
    float hj = p * d1W[j] + d1b[j];
    hj = hj > 0.f ? hj : 0.f;
    o += hj * d2W[j];
  }
  out[t] = o;
}

// ------------------------------ host helpers -------------------------------
static inline void launch_gemm(hipStream_t st, const float* A, int lda, long sAb,
                               const float* W, int ldw, long sWb, const float* bias,
                               float* C, int ldc, long sCb,
                               int M, int N, int K, int batch, int act) {
  if (N % 64 == 0) {
    dim3 g(M / 16, N / 64, batch);
    gemm_wmma_f32<4><<<g, 32, 0, st>>>(A, lda, sAb, W, ldw, sWb, bias, C, ldc, sCb, K, act);
  } else if (N % 32 == 0) {
    dim3 g(M / 16, N / 32, batch);
    gemm_wmma_f32<2><<<g, 32, 0, st>>>(A, lda, sAb, W, ldw, sWb, bias, C, ldc, sCb, K, act);
  } else {
    dim3 g(M / 16, N / 16, batch);
    gemm_wmma_f32<1><<<g, 32, 0, st>>>(A, lda, sAb, W, ldw, sWb, bias, C, ldc, sCb, K, act);
  }
}

static inline long gE(long n) { return (n + 255) / 256; }

// workspace layout (float offsets)
#define ALLOCF(name, count) constexpr long name = _off_##name; \
  constexpr long _end_##name = _off_##name + (count);
constexpr long _off_F_HNA = 0;                 ALLOCF(F_HNA, (long)NNODE * HID)
constexpr long _off_F_HNB = _end_F_HNA;        ALLOCF(F_HNB, (long)NNODE * HID)
constexpr long _off_F_HTA = _end_F_HNB;        ALLOCF(F_HTA, (long)NTOK * HID)
constexpr long _off_F_HTB = _end_F_HTA;        ALLOCF(F_HTB, (long)NTOK * HID)
constexpr long _off_F_K   = _end_F_HTB;        ALLOCF(F_K,   (long)NNODE * HID)
constexpr long _off_F_Q   = _end_F_K;          ALLOCF(F_Q,   (long)NNODE * HID)
constexpr long _off_F_V   = _end_F_Q;          ALLOCF(F_V,   (long)NNODE * HID)
constexpr long _off_F_QT  = _end_F_V;          ALLOCF(F_QT,  (long)NTOK * HID)
constexpr long _off_F_KR  = _end_F_QT;         ALLOCF(F_KR,  (long)NNODE * HID)
constexpr long _off_F_VR  = _end_F_KR;         ALLOCF(F_VR,  (long)NNODE * HID)
constexpr long _off_F_ON  = _end_F_VR;         ALLOCF(F_ON,  (long)NNODE * HID)
constexpr long _off_F_OT  = _end_F_ON;         ALLOCF(F_OT,  (long)NTOK * HID)
constexpr long _off_F_AMN = _end_F_OT;         ALLOCF(F_AMN, (long)NNODE * NHEAD)
constexpr long _off_F_DNN = _end_F_AMN;        ALLOCF(F_DNN, (long)NNODE * NHEAD)
constexpr long _off_F_ALN = _end_F_DNN;        ALLOCF(F_ALN, (long)ENN * NHEAD)
constexpr long _off_F_AMT = _end_F_ALN;        ALLOCF(F_AMT, (long)NTOK * NHEAD)
constexpr long _off_F_DNT = _end_F_AMT;        ALLOCF(F_DNT, (long)NTOK * NHEAD)
constexpr long _off_F_ALT = _end_F_DNT;        ALLOCF(F_ALT, (long)NNODE * NHEAD)
constexpr long _off_F_TE  = _end_F_ALT;        ALLOCF(F_TE,  (long)NTOK * HID)
constexpr long _off_F_CNT = _end_F_TE;         ALLOCF(F_CNT, (long)NTOK)
constexpr long _off_F_X   = _end_F_CNT;        ALLOCF(F_X,   (long)NTOK * HID)
constexpr long _off_F_QKV = _end_F_X;          ALLOCF(F_QKV, (long)NTOK * 384)
constexpr long _off_F_O   = _end_F_QKV;        ALLOCF(F_O,   (long)NTOK * HID)
constexpr long _off_F_OL  = _end_F_O;          ALLOCF(F_OL,  (long)NTOK * HID)
constexpr long _off_F_FF  = _end_F_OL;         ALLOCF(F_FF,  (long)NTOK * 512)

extern "C" void kernel_launch(void* const* d_in, const int* in_sizes, int n_in,
                              void* d_out, int out_size, void* d_ws, size_t ws_size,
                              hipStream_t stream) {
  (void)in_sizes; (void)n_in; (void)out_size; (void)ws_size;
  float* ws = (float*)d_ws;
#define PF(i) ((const float*)d_in[(i)])
  const int* enn = (const int*)d_in[2];  // [0:ENN]=src, [ENN:2ENN]=dst
  const int* ebt = (const int*)d_in[3];  // [0:NNODE]=src, [NNODE:2NNODE]=dst
  float* hn[2] = {ws + F_HNA, ws + F_HNB};
  float* ht[2] = {ws + F_HTA, ws + F_HTB};

  // ---- input projections ----
  launch_gemm(stream, PF(0), 16, 0, PF(4), HID, 0, PF(5), hn[0], HID, 0, NNODE, HID, 16, 1, 0);
  launch_gemm(stream, PF(1),  8, 0, PF(6), HID, 0, PF(7), ht[0], HID, 0, NTOK,  HID,  8, 1, 0);

  // ---- two HGT conv layers ----
  for (int L = 0; L < 2; ++L) {
    const int base = 8 + 24 * L;
    const float* hin = hn[L & 1];
    const float* tin = ht[L & 1];
    float* hout = hn[(L + 1) & 1];
    float* tout = ht[(L + 1) & 1];

    // K/Q/V on nodes, Q on time
    launch_gemm(stream, hin, HID, 0, PF(base + 0), HID, 0, PF(base + 1), ws + F_K, HID, 0, NNODE, HID, HID, 1, 0);
    launch_gemm(stream, hin, HID, 0, PF(base + 4), HID, 0, PF(base + 5), ws + F_Q, HID, 0, NNODE, HID, HID, 1, 0);
    launch_gemm(stream, hin, HID, 0, PF(base + 8), HID, 0, PF(base + 9), ws + F_V, HID, 0, NNODE, HID, HID, 1, 0);
    launch_gemm(stream, tin, HID, 0, PF(base + 6), HID, 0, PF(base + 7), ws + F_QT, HID, 0, NTOK, HID, HID, 1, 0);

    // relation nn: krel/vrel (batched per-head 32x32), then segment softmax
    launch_gemm(stream, ws + F_K, HID, DHEAD, PF(base + 18), DHEAD, DHEAD * DHEAD, nullptr,
                ws + F_KR, HID, DHEAD, NNODE, DHEAD, DHEAD, NHEAD, 0);
    launch_gemm(stream, ws + F_V, HID, DHEAD, PF(base + 19), DHEAD, DHEAD * DHEAD, nullptr,
                ws + F_VR, HID, DHEAD, NNODE, DHEAD, DHEAD, NHEAD, 0);
    fill_f32<<<gE((long)NNODE * NHEAD), 256, 0, stream>>>(ws + F_AMN, -INFINITY, (long)NNODE * NHEAD);
    fill_f32<<<gE((long)NNODE * NHEAD), 256, 0, stream>>>(ws + F_DNN, 0.f, (long)NNODE * NHEAD);
    fill_f32<<<gE((long)NNODE * HID),   256, 0, stream>>>(ws + F_ON,  0.f, (long)NNODE * HID);
    edge_alpha <<<gE((long)ENN * NHEAD), 256, 0, stream>>>(enn, enn + ENN, ws + F_Q, ws + F_KR,
                                                           PF(base + 20), ws + F_ALN, ws + F_AMN, ENN);
    edge_expsum<<<gE((long)ENN * NHEAD), 256, 0, stream>>>(enn + ENN, ws + F_ALN, ws + F_AMN, ws + F_DNN, ENN);
    edge_agg   <<<gE((long)ENN * NHEAD), 256, 0, stream>>>(enn, enn + ENN, ws + F_ALN, ws + F_DNN,
                                                           ws + F_VR, ws + F_ON, ENN);

    // relation bt: reuse krel/vrel buffers
    launch_gemm(stream, ws + F_K, HID, DHEAD, PF(base + 21), DHEAD, DHEAD * DHEAD, nullptr,
                ws + F_KR, HID, DHEAD, NNODE, DHEAD, DHEAD, NHEAD, 0);
    launch_gemm(stream, ws + F_V, HID, DHEAD, PF(base + 22), DHEAD, DHEAD * DHEAD, nullptr,
                ws + F_VR, HID, DHEAD, NNODE, DHEAD, DHEAD, NHEAD, 0);
    fill_f32<<<gE((long)NTOK * NHEAD), 256, 0, stream>>>(ws + F_AMT, -INFINITY, (long)NTOK * NHEAD);
    fill_f32<<<gE((long)NTOK * NHEAD), 256, 0, stream>>>(ws + F_DNT, 0.f, (long)NTOK * NHEAD);
    fill_f32<<<gE((long)NTOK * HID),   256, 0, stream>>>(ws + F_OT,  0.f, (long)NTOK * HID);
    edge_alpha <<<gE((long)NNODE * NHEAD), 256, 0, stream>>>(ebt, ebt + NNODE, ws + F_QT, ws + F_KR,
                                                             PF(base + 23), ws + F_ALT, ws + F_AMT, NNODE);
    edge_expsum<<<gE((long)NNODE * NHEAD), 256, 0, stream>>>(ebt + NNODE, ws + F_ALT, ws + F_AMT, ws + F_DNT, NNODE);
    edge_agg   <<<gE((long)NNODE * NHEAD), 256, 0, stream>>>(ebt, ebt + NNODE, ws + F_ALT, ws + F_DNT,
                                                             ws + F_VR, ws + F_OT, NNODE);

    // gelu -> A-linear -> sigmoid-skip -> relu
    gelu_ip<<<gE((long)NNODE * HID), 256, 0, stream>>>(ws + F_ON, (long)NNODE * HID);
    gelu_ip<<<gE((long)NTOK * HID),  256, 0, stream>>>(ws + F_OT, (long)NTOK * HID);
    launch_gemm(stream, ws + F_ON, HID, 0, PF(base + 12), HID, 0, PF(base + 13),
                ws + F_K, HID, 0, NNODE, HID, HID, 1, 0);   // reuse F_K as o_n
    skip_relu<<<gE((long)NNODE * HID), 256, 0, stream>>>(ws + F_K, hin, PF(base + 16), hout, (long)NNODE * HID);
    launch_gemm(stream, ws + F_OT, HID, 0, PF(base + 14), HID, 0, PF(base + 15),
                ws + F_QT, HID, 0, NTOK, HID, HID, 1, 0);   // reuse F_QT as o_t
    skip_relu<<<gE((long)NTOK * HID), 256, 0, stream>>>(ws + F_QT, tin, PF(base + 17), tout, (long)NTOK * HID);
  }

  // ---- scatter-mean nodes -> time slots ----
  fill_f32<<<gE((long)NTOK * HID), 256, 0, stream>>>(ws + F_TE, 0.f, (long)NTOK * HID);
  fill_f32<<<gE((long)NTOK), 256, 0, stream>>>(ws + F_CNT, 0.f, (long)NTOK);
  scatter_sum<<<gE((long)NNODE * HID), 256, 0, stream>>>(ebt, ebt + NNODE, hn[0], ws + F_TE, ws + F_CNT, NNODE);
  scatter_div<<<gE((long)NTOK * HID), 256, 0, stream>>>(ws + F_TE, ws + F_CNT);

  // ---- time transformer ----
  launch_gemm(stream, ws + F_TE, HID, 0, PF(56), HID, 0, PF(57), ws + F_X, HID, 0, NTOK, HID, HID, 1, 0);
  for (int L = 0; L < 2; ++L) {
    const int e = 58 + 12 * L;
    launch_gemm(stream, ws + F_X, HID, 0, PF(e + 0), 384, 0, PF(e + 1), ws + F_QKV, 384, 0, NTOK, 384, HID, 1, 0);
    mha_enc<<<dim3(NTOK, NHEAD), 256, 0, stream>>>(ws + F_QKV, ws + F_O);
    launch_gemm(stream, ws + F_O, HID, 0, PF(e + 2), HID, 0, PF(e + 3), ws + F_OL, HID, 0, NTOK, HID, HID, 1, 0);
    ln_residual<<<NTOK, HID, 0, stream>>>(ws + F_X, ws + F_OL, PF(e + 8), PF(e + 9));
    launch_gemm(stream, ws + F_X, HID, 0, PF(e + 4), 512, 0, PF(e + 5), ws + F_FF, 512, 0, NTOK, 512, HID, 1, 1);
    launch_gemm(stream, ws + F_FF, 512, 0, PF(e + 6), HID, 0, PF(e + 7), ws + F_OL, HID, 0, NTOK, HID, 512, 1, 0);
    ln_residual<<<NTOK, HID, 0, stream>>>(ws + F_X, ws + F_OL, PF(e + 10), PF(e + 11));
  }

  // ---- head + decoder ----
  head_dec<<<NTOK / 256, 256, 0, stream>>>(ws + F_X, PF(82), PF(83), PF(84), PF(85), PF(86), PF(87),
                                           (float*)d_out);
}